// DGLJTNNDecoder_7009386627277
// MI455X (gfx1250) — compile-verified
//
#include <hip/hip_runtime.h>
#include <hip/hip_bf16.h>
#include <math.h>

static constexpr int Hc  = 450;
static constexpr int LATc = 56;
static constexpr int Vc  = 780;
static constexpr int Tc  = 1024;   // power of two: row>>10 / row&1023
static constexpr int Lc  = 24;
static constexpr int NFc = 23;
static constexpr int NEc = 46;
static constexpr int QMc = Tc * Lc;        // 24576
static constexpr int PMc = Tc * (NEc + 1); // 48128
static constexpr int BM = 64, BN = 128, KC = 32;

typedef __attribute__((ext_vector_type(2))) float v2f;
typedef __attribute__((ext_vector_type(8))) float v8f;

enum GemmMode { M_GZ = 0, M_GH, M_GR, M_Q1, M_Q2, M_P1 };

struct GArgs {
  const float* x;       // embedded tokens (T,L,H)
  const float* a1;      // 2nd-half A (prev_m / prev_rm / cur_m / plain A)
  const float* b0;      // weights (first K half)
  const float* b1;      // weights (second K half) or null
  const float* bias;
  float* out;
  const float* zbuf;    // GH: z values
  const float* sbuf;    // GH: s (prev_m) or null
  const float* hprev;   // GH: mF[i-1] or null
  float* hs_out;        // GH: hs + step*T*H
  float* mF_out;        // GH: mF + step*T*H (forward only) or null
  const float* curm;    // GR: m_new
  const float* tree;    // tree_vec (T,LAT)
  const float* hs_all;  // hs base (Q1/P1)
  int M, N, K, idx;     // idx = src or dst node
};

__device__ __forceinline__ float sigm(float v) { return 1.f / (1.f + expf(-v)); }
__device__ __forceinline__ float softplusf(float v) {
  return log1pf(expf(-fabsf(v))) + fmaxf(v, 0.f);
}

__device__ __forceinline__ v8f wmma4(v2f a, v2f b, v8f c) {
  return __builtin_amdgcn_wmma_f32_16x16x4_f32(false, a, false, b, (short)0, c,
                                               false, false);
}

template <int MODE>
__device__ __forceinline__ float aload(const GArgs& g, int m, int kg) {
  if (kg >= g.K) return 0.f;
  if constexpr (MODE == M_GZ || MODE == M_GH || MODE == M_GR) {
    int t = m;
    if (kg < Hc) return g.x[((size_t)t * Lc + g.idx) * Hc + kg];
    return g.a1 ? g.a1[(size_t)t * Hc + (kg - Hc)] : 0.f;
  } else if constexpr (MODE == M_Q1) {
    int l = m >> 10, t = m & 1023;
    if (kg < Hc)
      return l == 0 ? 0.f : g.hs_all[(size_t)(l - 1) * Tc * Hc + (size_t)t * Hc + kg];
    return g.tree[(size_t)t * LATc + (kg - Hc)];
  } else if constexpr (MODE == M_Q2) {
    return g.a1[(size_t)m * Hc + kg];
  } else {  // M_P1
    int l = m >> 10, t = m & 1023;
    if (kg < Hc) {
      int node = 0;
      if (l > 0) {
        int st = l - 1;
        node = (st < NFc) ? st + 1 : 2 * NFc - 1 - st;  // dst of step st
      }
      return g.x[((size_t)t * Lc + node) * Hc + kg];
    }
    if (kg < 2 * Hc)
      return l == 0 ? 0.f
                    : g.hs_all[(size_t)(l - 1) * Tc * Hc + (size_t)t * Hc + (kg - Hc)];
    return g.tree[(size_t)t * LATc + (kg - 2 * Hc)];
  }
}

template <int MODE>
__device__ __forceinline__ float bload(const GArgs& g, int kg, int n) {
  if (kg >= g.K || n >= g.N) return 0.f;
  if constexpr (MODE == M_GZ || MODE == M_GH || MODE == M_GR) {
    if (kg < Hc) return g.b0[(size_t)kg * Hc + n];
    return g.b1[(size_t)(kg - Hc) * Hc + n];
  } else if constexpr (MODE == M_Q2) {
    return g.b0[(size_t)kg * Vc + n];
  } else {  // Q1 / P1: (K,450) row-major
    return g.b0[(size_t)kg * Hc + n];
  }
}

template <int MODE>
__device__ __forceinline__ void epilog(const GArgs& g, int m, int n, float acc) {
  if constexpr (MODE == M_GZ) {
    g.out[(size_t)m * Hc + n] = sigm(acc + g.bias[n]);
  } else if constexpr (MODE == M_GH) {
    float mt = tanhf(acc + g.bias[n]);
    float z = g.zbuf[(size_t)m * Hc + n];
    float s = g.sbuf ? g.sbuf[(size_t)m * Hc + n] : 0.f;
    float mn = (1.f - z) * s + z * mt;
    g.out[(size_t)m * Hc + n] = mn;  // cur_m
    float h = mn + (g.hprev ? g.hprev[(size_t)m * Hc + n] : 0.f);
    g.hs_out[(size_t)m * Hc + n] = h;
    if (g.mF_out) g.mF_out[(size_t)m * Hc + n] = mn;
  } else if constexpr (MODE == M_GR) {
    float r = sigm(acc + g.bias[n]);
    g.out[(size_t)m * Hc + n] = r * g.curm[(size_t)m * Hc + n];  // rm
  } else if constexpr (MODE == M_Q1 || MODE == M_P1) {
    float v = acc + g.bias[n];
    g.out[(size_t)m * Hc + n] = v > 0.f ? v : 0.f;
  } else {  // Q2
    g.out[(size_t)m * Vc + n] = acc + g.bias[n];
  }
}

// 256 threads = 8 wave32s; block tile 64x128; each wave a 32x32 C block
// (2x2 register tiles -> 4 independent fp32 WMMA accumulation chains).
// Double-buffered LDS: one barrier per K chunk, global fills overlap wmma.
template <int MODE>
__global__ __launch_bounds__(256) void gemm_k(GArgs g) {
  __shared__ float As[2][BM][KC + 1];
  __shared__ float Bs[2][KC][BN];
  const int tid = threadIdx.x;
  const int lane = tid & 31;
  const int wave = tid >> 5;
  const int wm = (wave >> 2) * 32;  // 0 or 32
  const int wn = (wave & 3) * 32;   // 0..96
  const int blockM = blockIdx.y * BM;
  const int blockN = blockIdx.x * BN;
  const int row = lane & 15;          // A row / C column lane
  const int koff = (lane >> 4) << 1;  // 0 or 2 (fp32 WMMA k split)

  auto fill = [&](int buf, int kbase) {
#pragma unroll
    for (int e = 0; e < (BM * KC) / 256; ++e) {
      int idx = tid + e * 256;
      int ar = idx >> 5, ak = idx & (KC - 1);
      As[buf][ar][ak] = aload<MODE>(g, blockM + ar, kbase + ak);
    }
#pragma unroll
    for (int e = 0; e < (KC * BN) / 256; ++e) {
      int idx = tid + e * 256;
      int bk = idx >> 7, bn = idx & (BN - 1);
      Bs[buf][bk][bn] = bload<MODE>(g, kbase + bk, blockN + bn);
    }
  };

  v8f acc00 = {}, acc01 = {}, acc10 = {}, acc11 = {};

  fill(0, 0);
  __syncthreads();
  int cur = 0;
  for (int k0 = 0; k0 < g.K; k0 += KC) {
    if (k0 + KC < g.K) fill(cur ^ 1, k0 + KC);
#pragma unroll
    for (int ks = 0; ks < KC / 4; ++ks) {
      const int kb = ks * 4 + koff;
      v2f a0, a1, b0, b1;
      a0.x = As[cur][wm + row][kb];
      a0.y = As[cur][wm + row][kb + 1];
      a1.x = As[cur][wm + 16 + row][kb];
      a1.y = As[cur][wm + 16 + row][kb + 1];
      b0.x = Bs[cur][kb][wn + row];
      b0.y = Bs[cur][kb + 1][wn + row];
      b1.x = Bs[cur][kb][wn + 16 + row];
      b1.y = Bs[cur][kb + 1][wn + 16 + row];
      acc00 = wmma4(a0, b0, acc00);
      acc01 = wmma4(a0, b1, acc01);
      acc10 = wmma4(a1, b0, acc10);
      acc11 = wmma4(a1, b1, acc11);
    }
    __syncthreads();
    cur ^= 1;
  }

  auto store_tile = [&](int moff, int noff, v8f acc) {
#pragma unroll
    for (int r = 0; r < 8; ++r) {
      int mrow = blockM + wm + moff + ((lane < 16) ? r : 8 + r);
      int ncol = blockN + wn + noff + row;
      if (ncol < g.N) epilog<MODE>(g, mrow, ncol, acc[r]);
    }
  };
  store_tile(0, 0, acc00);
  store_tile(0, 16, acc01);
  store_tile(16, 0, acc10);
  store_tile(16, 16, acc11);
}

__global__ void embed_k(const int* __restrict__ wid, const float* __restrict__ emb,
                        float* __restrict__ x) {
  size_t total = (size_t)Tc * Lc * Hc;
  for (size_t i = (size_t)blockIdx.x * blockDim.x + threadIdx.x; i < total;
       i += (size_t)gridDim.x * blockDim.x) {
    size_t tl = i / Hc;
    int h = (int)(i % Hc);
    x[i] = emb[(size_t)wid[tl] * Hc + h];
  }
}

__global__ void plogit_k(const float* __restrict__ phid, const float* __restrict__ usw,
                         const float* __restrict__ usb, float* __restrict__ plog) {
  int wave = threadIdx.x >> 5, lane = threadIdx.x & 31;
  int r = blockIdx.x * 8 + wave;
  if (r >= PMc) return;
  float sum = 0.f;
  for (int k = lane; k < Hc; k += 32) sum += phid[(size_t)r * Hc + k] * usw[k];
  for (int o = 16; o; o >>= 1) sum += __shfl_down(sum, o, 32);
  if (lane == 0) plog[r] = sum + usb[0];
}

__global__ void init_k(float* accum) {
  if (threadIdx.x < 4) accum[threadIdx.x] = 0.f;
}

__global__ __launch_bounds__(256) void qloss_k(const float* __restrict__ qlog,
                                               const int* __restrict__ wid,
                                               float* accum) {
  __shared__ float smax[256];
  __shared__ int sidx[256];
  __shared__ float ssum[256];
  const int r = blockIdx.x;
  const float* lrow = qlog + (size_t)r * Vc;
  float mx = -INFINITY;
  int mi = 0;
  for (int j = threadIdx.x; j < Vc; j += 256) {
    float v = lrow[j];
    if (v > mx) { mx = v; mi = j; }
  }
  smax[threadIdx.x] = mx;
  sidx[threadIdx.x] = mi;
  __syncthreads();
  for (int s = 128; s; s >>= 1) {
    if (threadIdx.x < s) {
      float v = smax[threadIdx.x + s];
      int i2 = sidx[threadIdx.x + s];
      if (v > smax[threadIdx.x] || (v == smax[threadIdx.x] && i2 < sidx[threadIdx.x])) {
        smax[threadIdx.x] = v;
        sidx[threadIdx.x] = i2;
      }
    }
    __syncthreads();
  }
  const float gmax = smax[0];
  const int gargmax = sidx[0];
  float se = 0.f;
  for (int j = threadIdx.x; j < Vc; j += 256) se += expf(lrow[j] - gmax);
  ssum[threadIdx.x] = se;
  __syncthreads();
  for (int s = 128; s; s >>= 1) {
    if (threadIdx.x < s) ssum[threadIdx.x] += ssum[threadIdx.x + s];
    __syncthreads();
  }
  if (threadIdx.x == 0) {
    int l = r >> 10, t = r & 1023;
    int tgt = wid[(size_t)t * Lc + l];
    float lse = gmax + logf(ssum[0]);
    atomicAdd(&accum[0], lse - lrow[tgt]);
    atomicAdd(&accum[2], (gargmax == tgt) ? 1.f : 0.f);
  }
}

__global__ __launch_bounds__(256) void ploss_k(const float* __restrict__ plog,
                                               float* accum) {
  __shared__ float sl[256], sa[256];
  int r = blockIdx.x * 256 + threadIdx.x;
  float loss = 0.f, ac = 0.f;
  if (r < PMc) {
    float xv = plog[r];
    float tgt = (r < NEc * Tc) ? (((r >> 10) < NFc) ? 1.f : 0.f) : 0.f;
    loss = tgt * softplusf(-xv) + (1.f - tgt) * softplusf(xv);
    ac = (((xv > 0.f) ? 1 : 0) == (int)tgt) ? 1.f : 0.f;
  }
  sl[threadIdx.x] = loss;
  sa[threadIdx.x] = ac;
  __syncthreads();
  for (int s = 128; s; s >>= 1) {
    if (threadIdx.x < s) {
      sl[threadIdx.x] += sl[threadIdx.x + s];
      sa[threadIdx.x] += sa[threadIdx.x + s];
    }
    __syncthreads();
  }
  if (threadIdx.x == 0) {
    atomicAdd(&accum[1], sl[0]);
    atomicAdd(&accum[3], sa[0]);
  }
}

__global__ void finalize_k(const float* accum, float* out) {
  if (threadIdx.x == 0) {
    out[0] = accum[0] / (float)Tc;   // q_loss
    out[1] = accum[1] / (float)Tc;   // p_loss
    out[2] = accum[2] / (float)QMc;  // q_acc
    out[3] = accum[3] / (float)PMc;  // p_acc
  }
}

extern "C" void kernel_launch(void* const* d_in, const int* in_sizes, int n_in,
                              void* d_out, int out_size, void* d_ws, size_t ws_size,
                              hipStream_t stream) {
  const int* wid = (const int*)d_in[0];
  const float* tree_vec = (const float*)d_in[1];
  const float* emb = (const float*)d_in[2];
  const float* W_w = (const float*)d_in[3];
  const float* W_b = (const float*)d_in[4];
  const float* U_w = (const float*)d_in[5];
  const float* U_b = (const float*)d_in[6];
  const float* Wo_w = (const float*)d_in[7];
  const float* Wo_b = (const float*)d_in[8];
  const float* Us_w = (const float*)d_in[9];
  const float* Us_b = (const float*)d_in[10];
  const float* Wz_w = (const float*)d_in[11];
  const float* Wz_b = (const float*)d_in[12];
  const float* Wr_w = (const float*)d_in[13];
  const float* Ur_w = (const float*)d_in[14];
  const float* Ur_b = (const float*)d_in[15];
  const float* Wh_w = (const float*)d_in[16];
  const float* Wh_b = (const float*)d_in[17];

  float* ws = (float*)d_ws;
  size_t off = 0;
  auto alloc = [&](size_t n) { float* p = ws + off; off += n; return p; };
  float* x    = alloc((size_t)Tc * Lc * Hc);
  float* mF   = alloc((size_t)NFc * Tc * Hc);
  float* hs   = alloc((size_t)NEc * Tc * Hc);
  float* mb0  = alloc((size_t)Tc * Hc);
  float* mb1  = alloc((size_t)Tc * Hc);
  float* rb0  = alloc((size_t)Tc * Hc);
  float* rb1  = alloc((size_t)Tc * Hc);
  float* zb   = alloc((size_t)Tc * Hc);
  float* qhid = alloc((size_t)QMc * Hc);
  float* qlog = alloc((size_t)QMc * Vc);
  float* phid = alloc((size_t)PMc * Hc);
  float* plog = alloc((size_t)PMc);
  float* accum = alloc(4);

  embed_k<<<2048, 256, 0, stream>>>(wid, emb, x);
  init_k<<<1, 32, 0, stream>>>(accum);

  const dim3 stepGrid((Hc + BN - 1) / BN, Tc / BM);
  for (int s = 0; s < NEc; ++s) {
    const bool fwd = s < NFc;
    const int i = fwd ? s : (2 * NFc - 1 - s);
    const int src = fwd ? i : i + 1;
    const int dst = fwd ? i + 1 : i;
    const bool have_pred = (s != 0 && s != NFc);
    float* cur_m  = (s & 1) ? mb1 : mb0;
    float* prev_m = (s & 1) ? mb0 : mb1;
    float* cur_rm  = (s & 1) ? rb1 : rb0;
    float* prev_rm = (s & 1) ? rb0 : rb1;

    GArgs gz{};
    gz.x = x; gz.a1 = have_pred ? prev_m : nullptr;
    gz.b0 = Wz_w; gz.b1 = Wz_w + (size_t)Hc * Hc; gz.bias = Wz_b; gz.out = zb;
    gz.M = Tc; gz.N = Hc; gz.K = 2 * Hc; gz.idx = src;
    gemm_k<M_GZ><<<stepGrid, 256, 0, stream>>>(gz);

    GArgs gh{};
    gh.x = x; gh.a1 = have_pred ? prev_rm : nullptr;
    gh.b0 = Wh_w; gh.b1 = Wh_w + (size_t)Hc * Hc; gh.bias = Wh_b; gh.out = cur_m;
    gh.zbuf = zb; gh.sbuf = have_pred ? prev_m : nullptr;
    gh.hprev = (!fwd && i > 0) ? (mF + (size_t)(i - 1) * Tc * Hc) : nullptr;
    gh.hs_out = hs + (size_t)s * Tc * Hc;
    gh.mF_out = fwd ? (mF + (size_t)s * Tc * Hc) : nullptr;
    gh.M = Tc; gh.N = Hc; gh.K = 2 * Hc; gh.idx = src;
    gemm_k<M_GH><<<stepGrid, 256, 0, stream>>>(gh);

    GArgs gr{};
    gr.x = x; gr.a1 = cur_m; gr.b0 = Wr_w; gr.b1 = Ur_w;
    gr.bias = Ur_b; gr.out = cur_rm; gr.curm = cur_m;
    gr.M = Tc; gr.N = Hc; gr.K = 2 * Hc; gr.idx = dst;
    gemm_k<M_GR><<<stepGrid, 256, 0, stream>>>(gr);
  }

  GArgs q1{};
  q1.hs_all = hs; q1.tree = tree_vec; q1.b0 = W_w; q1.bias = W_b; q1.out = qhid;
  q1.M = QMc; q1.N = Hc; q1.K = Hc + LATc;
  gemm_k<M_Q1><<<dim3((Hc + BN - 1) / BN, QMc / BM), 256, 0, stream>>>(q1);

  GArgs q2{};
  q2.a1 = qhid; q2.b0 = Wo_w; q2.bias = Wo_b; q2.out = qlog;
  q2.M = QMc; q2.N = Vc; q2.K = Hc;
  gemm_k<M_Q2><<<dim3((Vc + BN - 1) / BN, QMc / BM), 256, 0, stream>>>(q2);

  GArgs p1{};
  p1.x = x; p1.hs_all = hs; p1.tree = tree_vec; p1.b0 = U_w; p1.bias = U_b; p1.out = phid;
  p1.M = PMc; p1.N = Hc; p1.K = 2 * Hc + LATc;
  gemm_k<M_P1><<<dim3((Hc + BN - 1) / BN, PMc / BM), 256, 0, stream>>>(p1);

  plogit_k<<<PMc / 8, 256, 0, stream>>>(phid, Us_w, Us_b, plog);
  qloss_k<<<QMc, 256, 0, stream>>>(qlog, wid, accum);
  ploss_k<<<(PMc + 255) / 256, 256, 0, stream>>>(plog, accum);
  finalize_k<<<1, 1, 0, stream>>>(accum, (float*)d_out);
}